// simple_GCN_res_regression_14912126452009
// MI455X (gfx1250) — compile-verified
//
#include <hip/hip_runtime.h>
#include <hip/hip_bf16.h>

#define HDIM 128
#define EPS_BN 1e-5f

typedef __attribute__((ext_vector_type(2))) float v2f;
typedef __attribute__((ext_vector_type(8))) float v8f;

// ---------------------------------------------------------------- utilities
__global__ __launch_bounds__(256) void k_zero(float* p, long long n) {
    long long i = (long long)blockIdx.x * blockDim.x + threadIdx.x;
    long long stride = (long long)gridDim.x * blockDim.x;
    for (; i < n; i += stride) p[i] = 0.0f;
}

__global__ __launch_bounds__(256) void k_deg_scatter(const int* __restrict__ dst,
                                                     float* __restrict__ deg, int E) {
    int t = blockIdx.x * 256 + threadIdx.x;
    if (t < E) atomicAdd(&deg[dst[t]], 1.0f);
}

__global__ __launch_bounds__(256) void k_dinv(float* __restrict__ deg, int n) {
    int t = blockIdx.x * 256 + threadIdx.x;
    if (t < n) deg[t] = rsqrtf(deg[t] + 1.0f);
}

// ------------------------------------------------- layer 1: rank-1 "GEMM"
__global__ __launch_bounds__(256) void k_rank1(const float* __restrict__ X,
                                               const float* __restrict__ W1,
                                               float* __restrict__ h, int n) {
    int t = blockIdx.x * 256 + threadIdx.x;        // over n*32 float4 chunks
    if (t >= n * 32) return;
    int row = t >> 5, c4 = (t & 31) << 2;
    float xv = X[row];
    const float4 w = *(const float4*)(W1 + c4);
    float4 o; o.x = xv * w.x; o.y = xv * w.y; o.z = xv * w.z; o.w = xv * w.w;
    *(float4*)(h + (long long)row * HDIM + c4) = o;
}

// ------------------------------------------- fp32 WMMA GEMM: h = x @ W
// x: [n x 128] row-major, W: [128 x 128] row-major, h: [n x 128]
// block = 256 threads (8 waves); block handles 32 rows.
// wave w: rowHalf = w&1 (16-row strip), colPair = w>>1 -> 32-col strip
// (two 16x16 WMMA tiles sharing each A fragment -> 2 WMMAs per ds_load).
__global__ __launch_bounds__(256) void k_gemm_wmma(const float* __restrict__ x,
                                                   const float* __restrict__ W,
                                                   float* __restrict__ h, int n) {
    __shared__ float tileX[32 * HDIM];             // 16 KB activation tile
    const int rowBase = blockIdx.x * 32;
    const int tid  = threadIdx.x;
    const int wave = tid >> 5;
    const int lane = tid & 31;
    const int half = lane >> 4;                    // 0|1 (K pair / M+8 select)
    const int l16  = lane & 15;
    const int rowHalf = wave & 1;
    const int c0   = (wave >> 1) * 32;

    const bool fullTile = (rowBase + 32 <= n);

    // stage 32x128 activation tile into LDS (1024 float4, 4 per thread)
    if (fullTile) {
        const float4* xs = (const float4*)(x + (long long)rowBase * HDIM);
        #pragma unroll
        for (int i = 0; i < 4; ++i) {
            int idx = tid + i * 256;
            ((float4*)tileX)[idx] = xs[idx];
        }
    } else {
        #pragma unroll
        for (int i = 0; i < 4; ++i) {
            int idx = tid + i * 256;
            int r = rowBase + (idx >> 5);
            float4 v = make_float4(0.f, 0.f, 0.f, 0.f);
            if (r < n) v = *(const float4*)(x + (long long)r * HDIM + ((idx & 31) << 2));
            ((float4*)tileX)[idx] = v;
        }
    }
    __syncthreads();

    // A frag base: lane<16 reads {k,k+1}, lane>=16 reads {k+2,k+3} of its row
    const float* xrow = tileX + (rowHalf * 16 + l16) * HDIM + 2 * half;
    v8f acc0 = {}, acc1 = {};

    #pragma unroll 4
    for (int k = 0; k < HDIM; k += 4) {
        v2f a; a.x = xrow[k]; a.y = xrow[k + 1];
        const float* wp = W + (long long)(k + 2 * half) * HDIM + c0 + l16;
        v2f b0; b0.x = wp[0];  b0.y = wp[HDIM];
        v2f b1; b1.x = wp[16]; b1.y = wp[HDIM + 16];
        acc0 = __builtin_amdgcn_wmma_f32_16x16x4_f32(false, a, false, b0,
                                                     (short)0, acc0, false, false);
        acc1 = __builtin_amdgcn_wmma_f32_16x16x4_f32(false, a, false, b1,
                                                     (short)0, acc1, false, false);
    }

    // D layout: VGPR v holds M = v + 8*half, N = l16
    const int rbase = rowBase + rowHalf * 16 + half * 8;
    if (fullTile) {
        #pragma unroll
        for (int v = 0; v < 8; ++v) {
            float* op = h + (long long)(rbase + v) * HDIM + c0 + l16;
            op[0]  = acc0[v];
            op[16] = acc1[v];
        }
    } else {
        for (int v = 0; v < 8; ++v) {
            int r = rbase + v;
            if (r < n) {
                h[(long long)r * HDIM + c0 + l16]      = acc0[v];
                h[(long long)r * HDIM + c0 + 16 + l16] = acc1[v];
            }
        }
    }
}

// ------------------------------------------------ edge scatter aggregation
__global__ __launch_bounds__(256) void k_edge_scatter(const float* __restrict__ h,
                                                      const int* __restrict__ src,
                                                      const int* __restrict__ dst,
                                                      const float* __restrict__ dinv,
                                                      float* __restrict__ agg, int E) {
    int t = blockIdx.x * 256 + threadIdx.x;        // E*32 threads (4 ch each)
    if (t >= E * 32) return;
    int e = t >> 5, c4 = (t & 31) << 2;
    int s = src[e], d = dst[e];
    float coef = dinv[s] * dinv[d];
    const float4 hv = *(const float4*)(h + (long long)s * HDIM + c4);
    float* ap = agg + (long long)d * HDIM + c4;
    atomicAdd(ap + 0, hv.x * coef);
    atomicAdd(ap + 1, hv.y * coef);
    atomicAdd(ap + 2, hv.z * coef);
    atomicAdd(ap + 3, hv.w * coef);
}

// -------------------------------- self-loop + bias + ReLU (in place on agg)
__global__ __launch_bounds__(256) void k_combine_relu(float* __restrict__ agg,
                                                      const float* __restrict__ h,
                                                      const float* __restrict__ dinv,
                                                      const float* __restrict__ bias, int n) {
    int t = blockIdx.x * 256 + threadIdx.x;
    if (t >= n * 32) return;
    int row = t >> 5, c4 = (t & 31) << 2;
    float di = dinv[row], d2 = di * di;
    long long off = (long long)row * HDIM + c4;
    float4 a  = *(const float4*)(agg + off);
    float4 hv = *(const float4*)(h + off);
    float4 b  = *(const float4*)(bias + c4);
    a.x = fmaxf(a.x + hv.x * d2 + b.x, 0.f);
    a.y = fmaxf(a.y + hv.y * d2 + b.y, 0.f);
    a.z = fmaxf(a.z + hv.z * d2 + b.z, 0.f);
    a.w = fmaxf(a.w + hv.w * d2 + b.w, 0.f);
    *(float4*)(agg + off) = a;
}

// --------------------------------------- BatchNorm stats (sum, sumsq) / ch
__global__ __launch_bounds__(256) void k_stats(const float* __restrict__ y,
                                               float* __restrict__ sums, int n) {
    int c = threadIdx.x & 127;
    int sub = threadIdx.x >> 7;                    // 0|1
    float s = 0.f, s2 = 0.f;
    for (int r = blockIdx.x * 2 + sub; r < n; r += gridDim.x * 2) {
        float v = y[(long long)r * HDIM + c];
        s += v; s2 += v * v;
    }
    __shared__ float ls[256], ls2[256];
    ls[threadIdx.x] = s; ls2[threadIdx.x] = s2;
    __syncthreads();
    if (sub == 0) {
        atomicAdd(&sums[c],        ls[c]  + ls[128 + c]);
        atomicAdd(&sums[128 + c],  ls2[c] + ls2[128 + c]);
    }
}

__global__ void k_scaleshift(const float* __restrict__ sums,
                             const float* __restrict__ gamma,
                             const float* __restrict__ beta,
                             float* __restrict__ ss, float invN) {
    int c = threadIdx.x;                            // 128 threads
    float m   = sums[c] * invN;
    float var = sums[128 + c] * invN - m * m;
    float sc  = rsqrtf(var + EPS_BN) * gamma[c];
    ss[c] = sc;
    ss[128 + c] = beta[c] - m * sc;
}

// ------------------------ BN apply (+ optional residual), in place on y
__global__ __launch_bounds__(256) void k_bn_apply(float* __restrict__ y,
                                                  const float* __restrict__ res,
                                                  const float* __restrict__ ss, int n) {
    int t = blockIdx.x * 256 + threadIdx.x;
    if (t >= n * 32) return;
    int row = t >> 5, c4 = (t & 31) << 2;
    long long off = (long long)row * HDIM + c4;
    float4 v  = *(const float4*)(y + off);
    float4 sc = *(const float4*)(ss + c4);
    float4 sh = *(const float4*)(ss + 128 + c4);
    v.x = v.x * sc.x + sh.x; v.y = v.y * sc.y + sh.y;
    v.z = v.z * sc.z + sh.z; v.w = v.w * sc.w + sh.w;
    if (res) {
        float4 r = *(const float4*)(res + off);
        v.x += r.x; v.y += r.y; v.z += r.z; v.w += r.w;
    }
    *(float4*)(y + off) = v;
}

// ---------------------------------------------------------- global pooling
__global__ __launch_bounds__(256) void k_counts(const int* __restrict__ batch,
                                                float* __restrict__ counts, int n) {
    int t = blockIdx.x * 256 + threadIdx.x;
    if (t < n) atomicAdd(&counts[batch[t]], 1.0f);
}

__global__ __launch_bounds__(256) void k_pool_acc(const float* __restrict__ x,
                                                  const int* __restrict__ batch,
                                                  float* __restrict__ pool, int n) {
    int t = blockIdx.x * 256 + threadIdx.x;
    if (t >= n * 32) return;
    int row = t >> 5, c4 = (t & 31) << 2;
    int g = batch[row];
    float4 v = *(const float4*)(x + (long long)row * HDIM + c4);
    float* p = pool + (long long)g * HDIM + c4;
    atomicAdd(p + 0, v.x); atomicAdd(p + 1, v.y);
    atomicAdd(p + 2, v.z); atomicAdd(p + 3, v.w);
}

__global__ __launch_bounds__(256) void k_pool_final(const float* __restrict__ pool,
                                                    const float* __restrict__ counts,
                                                    float* __restrict__ xmean, int G) {
    int t = blockIdx.x * 256 + threadIdx.x;
    if (t >= G * HDIM) return;
    int g = t / HDIM;
    xmean[t] = pool[t] / fmaxf(counts[g], 1.0f);
}

__global__ void k_pred(const float* __restrict__ xmean, const float* __restrict__ Wr,
                       const float* __restrict__ br, float* __restrict__ pred) {
    __shared__ float red[128];
    int g = blockIdx.x, c = threadIdx.x;
    red[c] = xmean[(long long)g * HDIM + c] * Wr[c];
    __syncthreads();
    for (int s = 64; s > 0; s >>= 1) {
        if (c < s) red[c] += red[c + s];
        __syncthreads();
    }
    if (c == 0) pred[g] = red[0] + br[0];
}

// ---------------------------------------------------------------- launcher
extern "C" void kernel_launch(void* const* d_in, const int* in_sizes, int n_in,
                              void* d_out, int out_size, void* d_ws, size_t ws_size,
                              hipStream_t stream) {
    const float* X     = (const float*)d_in[0];
    const int*   edge  = (const int*)d_in[1];
    const int*   batch = (const int*)d_in[2];
    const float* W1    = (const float*)d_in[3];
    const float* b1    = (const float*)d_in[4];
    const float* Wm    = (const float*)d_in[5];
    const float* bm    = (const float*)d_in[6];
    const float* gamma = (const float*)d_in[7];
    const float* beta  = (const float*)d_in[8];
    const float* Wr    = (const float*)d_in[9];
    const float* br    = (const float*)d_in[10];
    float* out = (float*)d_out;

    const int N = in_sizes[0];                 // 100000
    const int E = in_sizes[1] / 2;             // 1600000
    const int G = out_size / (HDIM + 1);       // 64
    const int* src = edge;
    const int* dst = edge + E;

    const long long NH = (long long)N * HDIM;
    float* ws    = (float*)d_ws;
    float* buf[3] = { ws, ws + NH, ws + 2 * NH };   // rotating N x H buffers
    float* dinv  = ws + 3 * NH;                     // N
    float* sums  = dinv + N;                        // 256
    float* ss    = sums + 256;                      // 256
    float* pool  = ss + 256;                        // G*H
    float* cnts  = pool + (long long)G * HDIM;      // G

    const int gE   = (E + 255) / 256;
    const int gN   = (N + 255) / 256;
    const int gNH4 = (N * 32 + 255) / 256;
    const int gE4  = (E * 32 + 255) / 256;
    const int gGem = (N + 31) / 32;

    // ---- degrees -> dinv
    k_zero<<<1024, 256, 0, stream>>>(dinv, N);
    k_deg_scatter<<<gE, 256, 0, stream>>>(dst, dinv, E);
    k_dinv<<<gN, 256, 0, stream>>>(dinv, N);

    // per-layer tail: scatter-agg -> combine/relu -> stats -> BN(+res)
    auto layer_tail = [&](float* hbuf, float* abuf, const float* bias,
                          const float* g_, const float* be_, const float* res) {
        k_zero<<<2048, 256, 0, stream>>>(abuf, NH);
        k_edge_scatter<<<gE4, 256, 0, stream>>>(hbuf, src, dst, dinv, abuf, E);
        k_combine_relu<<<gNH4, 256, 0, stream>>>(abuf, hbuf, dinv, bias, N);
        k_zero<<<1, 256, 0, stream>>>(sums, 256);
        k_stats<<<256, 256, 0, stream>>>(abuf, sums, N);
        k_scaleshift<<<1, 128, 0, stream>>>(sums, g_, be_, ss, 1.0f / (float)N);
        k_bn_apply<<<gNH4, 256, 0, stream>>>(abuf, res, ss, N);
    };

    // ---- layer 1 (rank-1 GEMM, no residual): result lands in buf[2]
    k_rank1<<<gNH4, 256, 0, stream>>>(X, W1, buf[1], N);
    layer_tail(buf[1], buf[2], b1, gamma, beta, nullptr);

    float* cur = buf[2];
    float* f0  = buf[0];
    float* f1  = buf[1];

    // ---- layers 2..5: WMMA GEMM + tail with residual; rotate buffers
    for (int i = 0; i < 4; ++i) {
        float* hbuf = f0;
        float* abuf = f1;
        k_gemm_wmma<<<gGem, 256, 0, stream>>>(cur, Wm + (long long)i * HDIM * HDIM, hbuf, N);
        layer_tail(hbuf, abuf, bm + (long long)i * HDIM,
                   gamma + (long long)(i + 1) * HDIM,
                   beta  + (long long)(i + 1) * HDIM, cur);
        float* next = abuf;      // result in abuf
        f1 = cur;                // old input now free
        f0 = hbuf;               // h buffer now free
        cur = next;
    }

    // ---- global mean pool + prediction head
    k_zero<<<32, 256, 0, stream>>>(pool, (long long)G * HDIM + G);
    k_counts<<<gN, 256, 0, stream>>>(batch, cnts, N);
    k_pool_acc<<<gNH4, 256, 0, stream>>>(cur, batch, pool, N);
    k_pool_final<<<(G * HDIM + 255) / 256, 256, 0, stream>>>(pool, cnts, out + G, G);
    k_pred<<<G, 128, 0, stream>>>(out + G, Wr, br, out);
}